// LPSparseMAP_50276887167515
// MI455X (gfx1250) — compile-verified
//
#include <hip/hip_runtime.h>
#include <hip/hip_bf16.h>
#include <math.h>

#define DIM       8192
#define BATCH     4096
#define NB_SPLIT  31
#define NB_NODES  63
#define KSPLIT    16
#define KCHUNK    (DIM / KSPLIT)   // 512

typedef float v2f __attribute__((ext_vector_type(2)));
typedef float v8f __attribute__((ext_vector_type(8)));

// ---------------------------------------------------------------------------
// Kernel 1: K-split partial GEMM  XA_partial[ks] = x[:, chunk] @ A[:, chunk]^T
// One wave computes a 16x32 output tile via V_WMMA_F32_16X16X4_F32.
// Branchless + software-pipelined: next fragments are loaded before the
// current WMMA pair so loads stay in flight across the matrix ops.
// ---------------------------------------------------------------------------
__global__ __launch_bounds__(32)
void gemm_xa_partial(const float* __restrict__ x, const float* __restrict__ A,
                     float* __restrict__ partial)
{
    const int lane  = threadIdx.x;     // 0..31
    const int m     = lane & 15;
    const int kHalf = lane >> 4;       // 0 or 1  (K pairs 0-1 vs 2-3)
    const int row0  = blockIdx.x * 16;
    const int kBase = blockIdx.y * KCHUNK;

    // Padded column 31: point lane 15's tile-1 row at A row 0 (valid memory).
    // Its contribution lands only in D column 31, which is never read.
    const int n1 = (16 + m < NB_SPLIT) ? (16 + m) : 0;

    const float* xp = x + (size_t)(row0 + m) * DIM + kBase + 2 * kHalf;
    const float* a0 = A + (size_t)m  * DIM + kBase + 2 * kHalf;
    const float* a1 = A + (size_t)n1 * DIM + kBase + 2 * kHalf;

    v8f c0 = {}; v8f c1 = {};

    // pipeline prologue: fragment for k-step 0
    v2f a_c  = *(const v2f*)(xp);
    v2f b0_c = *(const v2f*)(a0);
    v2f b1_c = *(const v2f*)(a1);

    // main loop: 15 groups of 8 k-steps, each step consumes cur & preloads next
    for (int kk = 0; kk < KCHUNK - 32; kk += 32) {
        __builtin_prefetch(xp + kk + 32, 0, 3);   // one 128B line ahead per lane stream
        #pragma unroll
        for (int u = 0; u < 8; ++u) {
            const int k = kk + 4 * u;
            v2f a_n  = *(const v2f*)(xp + k + 4);
            v2f b0_n = *(const v2f*)(a0 + k + 4);
            v2f b1_n = *(const v2f*)(a1 + k + 4);
            c0 = __builtin_amdgcn_wmma_f32_16x16x4_f32(false, a_c, false, b0_c, (short)0, c0, false, false);
            c1 = __builtin_amdgcn_wmma_f32_16x16x4_f32(false, a_c, false, b1_c, (short)0, c1, false, false);
            a_c = a_n; b0_c = b0_n; b1_c = b1_n;
        }
    }
    // epilogue group (last 8 k-steps, no load past the chunk end)
    #pragma unroll
    for (int u = 0; u < 8; ++u) {
        const int k = (KCHUNK - 32) + 4 * u;
        v2f a_n = a_c, b0_n = b0_c, b1_n = b1_c;
        if (u < 7) {
            a_n  = *(const v2f*)(xp + k + 4);
            b0_n = *(const v2f*)(a0 + k + 4);
            b1_n = *(const v2f*)(a1 + k + 4);
        }
        c0 = __builtin_amdgcn_wmma_f32_16x16x4_f32(false, a_c, false, b0_c, (short)0, c0, false, false);
        c1 = __builtin_amdgcn_wmma_f32_16x16x4_f32(false, a_c, false, b1_c, (short)0, c1, false, false);
        a_c = a_n; b0_c = b0_n; b1_c = b1_n;
    }

    // D layout: VGPR r => M=r (lanes 0-15) / M=8+r (lanes 16-31); N = lane&15
    float* p = partial + (size_t)blockIdx.y * BATCH * 32;
    #pragma unroll
    for (int r = 0; r < 8; ++r) {
        const int row = row0 + r + 8 * kHalf;
        p[(size_t)row * 32 + m]      = c0[r];
        p[(size_t)row * 32 + 16 + m] = c1[r];
    }
}

// ---------------------------------------------------------------------------
// Kernel 2: deterministic K-split reduction + exact tree-min propagation -> q
// ---------------------------------------------------------------------------
__global__ void reduce_tree(const float* __restrict__ partial, float* __restrict__ q)
{
    const int r = blockIdx.x * blockDim.x + threadIdx.x;
    if (r >= BATCH) return;

    float xa[NB_SPLIT];
    #pragma unroll
    for (int c = 0; c < NB_SPLIT; ++c) xa[c] = 0.f;
    for (int ks = 0; ks < KSPLIT; ++ks) {
        const float* p = partial + ((size_t)ks * BATCH + r) * 32;
        #pragma unroll
        for (int c = 0; c < NB_SPLIT; ++c) xa[c] += p[c];
    }

    // exact emulation of the vectorized .at[].set semantics (two-phase updates)
    float qq[NB_NODES];
    #pragma unroll
    for (int i = 0; i < NB_NODES; ++i) qq[i] = 1.f;
    float tl[NB_SPLIT], tr[NB_SPLIT];
    #pragma unroll
    for (int s = 0; s < NB_SPLIT; ++s) tl[s] = fminf(qq[s], xa[s]);
    #pragma unroll
    for (int s = 0; s < NB_SPLIT; ++s) qq[2 * s + 1] = tl[s];
    #pragma unroll
    for (int s = 0; s < NB_SPLIT; ++s) tr[s] = fminf(qq[s], -xa[s]);
    #pragma unroll
    for (int s = 0; s < NB_SPLIT; ++s) qq[2 * s + 2] = tr[s];
    for (int d = 0; d < 5; ++d) {
        #pragma unroll
        for (int s = 0; s < NB_SPLIT; ++s) tl[s] = fminf(qq[2 * s + 1], qq[s]);
        #pragma unroll
        for (int s = 0; s < NB_SPLIT; ++s) qq[2 * s + 1] = tl[s];
        #pragma unroll
        for (int s = 0; s < NB_SPLIT; ++s) tr[s] = fminf(qq[2 * s + 2], qq[s]);
        #pragma unroll
        for (int s = 0; s < NB_SPLIT; ++s) qq[2 * s + 2] = tr[s];
    }
    float* qr = q + (size_t)r * NB_NODES;
    #pragma unroll
    for (int i = 0; i < NB_NODES; ++i) qr[i] = qq[i];
}

// ---------------------------------------------------------------------------
// Kernel 3: per-node descending bitonic sort (LDS) + prefix sums + singleton d0
// ---------------------------------------------------------------------------
__global__ __launch_bounds__(1024)
void sort_node(const float* __restrict__ q, const float* __restrict__ eta,
               float* __restrict__ qs, float* __restrict__ cs, float* __restrict__ dvec)
{
    __shared__ float sv[BATCH];      // 16 KB
    __shared__ float ssum[1024];     // 4 KB
    __shared__ int   firstF;

    const int node = blockIdx.x;
    const int tid  = threadIdx.x;

    for (int e = tid; e < BATCH; e += 1024)
        sv[e] = q[(size_t)e * NB_NODES + node];
    if (tid == 0) firstF = BATCH;
    __syncthreads();

    // bitonic sort, descending
    for (int k = 2; k <= BATCH; k <<= 1) {
        for (int j = k >> 1; j > 0; j >>= 1) {
            for (int e = tid; e < BATCH; e += 1024) {
                const int partner = e ^ j;
                if (partner > e) {
                    const float a = sv[e], b = sv[partner];
                    const bool desc = ((e & k) == 0);
                    if (desc ? (a < b) : (a > b)) { sv[e] = b; sv[partner] = a; }
                }
            }
            __syncthreads();
        }
    }

    // local prefix over 4 owned elements + block scan of the 1024 partials
    float v[4], incl[4];
    float lsum = 0.f;
    #pragma unroll
    for (int i = 0; i < 4; ++i) { v[i] = sv[4 * tid + i]; lsum += v[i]; incl[i] = lsum; }
    ssum[tid] = lsum;
    __syncthreads();
    for (int off = 1; off < 1024; off <<= 1) {
        const float t = (tid >= off) ? ssum[tid - off] : 0.f;
        __syncthreads();
        ssum[tid] += t;
        __syncthreads();
    }
    const float base = (tid > 0) ? ssum[tid - 1] : 0.f;
    const float et   = eta[node];

    #pragma unroll
    for (int i = 0; i < 4; ++i) {
        const int   kidx = 4 * tid + i;
        const float csk  = incl[i] + base;
        const float prev = csk - v[i];
        const float dval = (et + prev) / (1.f + (float)kidx);
        if (v[i] < dval) atomicMin(&firstF, kidx);     // leading-run boundary
        qs[(size_t)node * BATCH + kidx] = v[i];
        cs[(size_t)node * BATCH + kidx] = csk;
    }
    __syncthreads();
    #pragma unroll
    for (int i = 0; i < 4; ++i) sv[4 * tid + i] = incl[i] + base;   // sv := cs
    __syncthreads();
    if (tid == 0) {
        const int   mcount = firstF;
        const float cm     = (mcount > 0) ? sv[mcount - 1] : 0.f;
        dvec[node] = (et + cm) / (1.f + (float)mcount);
    }
}

// ---------------------------------------------------------------------------
// Kernel 4: serial colored-merge scan (62 iters), single block of 64 threads.
// _colored_d solved as water-filling fixed point via bisection; per-column
// count/sum via binary search into sorted columns + prefix sums.
// ---------------------------------------------------------------------------
__global__ __launch_bounds__(64)
void scan_merge(const float* __restrict__ qs, const float* __restrict__ cs,
                const float* __restrict__ eta, float* __restrict__ dvec)
{
    __shared__ float sd[NB_NODES];
    __shared__ int   scol[NB_NODES];
    __shared__ float rf[64];
    __shared__ float rg[64];
    __shared__ int   ri[64];

    const int   tid = threadIdx.x;
    const float NEG = -3.0e38f;

    if (tid < NB_NODES) { sd[tid] = dvec[tid]; scol[tid] = tid; }
    __syncthreads();

    for (int iter = 0; iter < NB_NODES - 1; ++iter) {
        // --- find first max violating node t ---
        float myd = NEG;
        if (tid >= 1 && tid < NB_NODES) {
            const int par = (tid - 1) >> 1;
            if (sd[tid] > sd[par]) myd = sd[tid];
        }
        rf[tid] = myd; ri[tid] = tid;
        for (int off = 32; off > 0; off >>= 1) {
            __syncthreads();
            if (tid < off) {
                const float a = rf[tid], b = rf[tid + off];
                const int ia = ri[tid], ib = ri[tid + off];
                if (b > a || (b == a && ib < ia)) { rf[tid] = b; ri[tid] = ib; }
            }
        }
        __syncthreads();
        if (rf[0] == NEG) break;                  // no violation: done (uniform)
        const int t  = ri[0];
        const int p  = (t - 1) >> 1;
        const int pc = scol[p];
        __syncthreads();
        if (tid < NB_NODES && scol[tid] == t) scol[tid] = pc;   // merge color t -> pc
        __syncthreads();
        const bool inmask = (tid < NB_NODES) && (scol[tid] == pc);

        // --- n, S over the merged color class ---
        rf[tid] = inmask ? 1.f : 0.f;
        rg[tid] = inmask ? eta[tid] : 0.f;
        for (int off = 32; off > 0; off >>= 1) {
            __syncthreads();
            if (tid < off) { rf[tid] += rf[tid + off]; rg[tid] += rg[tid + off]; }
        }
        __syncthreads();
        const float n = rf[0], S = rg[0];

        // --- value range for bisection ---
        float mymin = 3.0e38f, mymax = NEG;
        if (inmask) {
            mymax = qs[(size_t)tid * BATCH];
            mymin = qs[(size_t)tid * BATCH + BATCH - 1];
        }
        __syncthreads();
        rf[tid] = mymin; rg[tid] = mymax;
        for (int off = 32; off > 0; off >>= 1) {
            __syncthreads();
            if (tid < off) { rf[tid] = fminf(rf[tid], rf[tid + off]); rg[tid] = fmaxf(rg[tid], rg[tid + off]); }
        }
        __syncthreads();
        const float ratio = S / n;
        float lo = fminf(rf[0], ratio) - 1.f;
        float hi = fmaxf(rg[0], ratio) + 1.f;

        // --- bisection: d = (S + sum(q>=d)) / (n + count(q>=d)) ---
        float cntF = 0.f, sumF = 0.f;
        for (int it = 0; it < 48; ++it) {
            const float mid = 0.5f * (lo + hi);
            float mycnt = 0.f, mysum = 0.f;
            if (inmask) {
                const float* col = qs + (size_t)tid * BATCH;
                int loI = 0, hiI = BATCH;
                while (loI < hiI) { const int mI = (loI + hiI) >> 1; if (col[mI] >= mid) loI = mI + 1; else hiI = mI; }
                mycnt = (float)loI;
                mysum = (loI > 0) ? cs[(size_t)tid * BATCH + loI - 1] : 0.f;
            }
            __syncthreads();
            rf[tid] = mycnt; rg[tid] = mysum;
            for (int off = 32; off > 0; off >>= 1) {
                __syncthreads();
                if (tid < off) { rf[tid] += rf[tid + off]; rg[tid] += rg[tid + off]; }
            }
            __syncthreads();
            cntF = rf[0]; sumF = rg[0];
            const float g = (S + sumF) / (n + cntF);
            if (g > mid) lo = mid; else hi = mid;   // uniform update
        }
        const float dm = (S + sumF) / (n + cntF);
        __syncthreads();
        if (inmask) sd[tid] = dm;
        __syncthreads();
    }
    __syncthreads();
    if (tid < NB_NODES) dvec[tid] = fminf(fmaxf(sd[tid], 0.f), 1.f);
}

// ---------------------------------------------------------------------------
// Kernel 5: z = min(clip(q,0,1), d[node])
// ---------------------------------------------------------------------------
__global__ void finalize(const float* __restrict__ q, const float* __restrict__ dvec,
                         float* __restrict__ out)
{
    const int i = blockIdx.x * blockDim.x + threadIdx.x;
    if (i >= BATCH * NB_NODES) return;
    const int node = i % NB_NODES;
    const float z = fminf(fmaxf(q[i], 0.f), 1.f);
    out[i] = fminf(z, dvec[node]);
}

// ---------------------------------------------------------------------------
extern "C" void kernel_launch(void* const* d_in, const int* in_sizes, int n_in,
                              void* d_out, int out_size, void* d_ws, size_t ws_size,
                              hipStream_t stream)
{
    const float* x   = (const float*)d_in[0];   // [4096, 8192]
    const float* A   = (const float*)d_in[1];   // [31, 8192]
    const float* eta = (const float*)d_in[2];   // [63]
    float* out = (float*)d_out;                 // [4096, 63]

    float* partial = (float*)d_ws;                                  // KSPLIT*BATCH*32
    float* q       = partial + (size_t)KSPLIT * BATCH * 32;         // BATCH*63
    float* qs      = q  + (size_t)BATCH * NB_NODES;                 // 63*BATCH
    float* csbuf   = qs + (size_t)NB_NODES * BATCH;                 // 63*BATCH
    float* dv      = csbuf + (size_t)NB_NODES * BATCH;              // 63

    dim3 g1(BATCH / 16, KSPLIT);
    gemm_xa_partial<<<g1, 32, 0, stream>>>(x, A, partial);
    reduce_tree<<<BATCH / 256, 256, 0, stream>>>(partial, q);
    sort_node<<<NB_NODES, 1024, 0, stream>>>(q, eta, qs, csbuf, dv);
    scan_merge<<<1, 64, 0, stream>>>(qs, csbuf, eta, dv);
    const int tot = BATCH * NB_NODES;
    finalize<<<(tot + 255) / 256, 256, 0, stream>>>(q, dv, out);
}